// GNNEncoder_31284541784160
// MI455X (gfx1250) — compile-verified
//
#include <hip/hip_runtime.h>
#include <hip/hip_bf16.h>
#include <math.h>

// ---------------------------------------------------------------------------
// GNN encoder layer for MI455X (gfx1250), fp32 via V_WMMA_F32_16X16X4_F32.
// B=2, Vsc=Vst=256, H=256.
// ---------------------------------------------------------------------------

typedef __attribute__((ext_vector_type(2))) float v2f;
typedef __attribute__((ext_vector_type(8))) float v8f;

#define WB 2
#define WV 256
#define WH 256
#define NROWS_H   (WB * WV)        // 512
#define EDGE_ROWS (WB * WV * WV)   // 131072
#define EPS_BN 1e-5f

// ---- workspace layout (float offsets) -------------------------------------
//  0 Uh_sc  1 Uh_st  2 Vh_sc  3 Vh_st  4 Wh_sc  5 Wh_st
//  6 A_bi(W6@h_sc)  7 B_bi(W7@h_st)  8 A_sc(W9@h_sc)  9 B_sc(W10@h_sc)
// 10 A_st(W12@h_st) 11 B_st(W13@h_st)
#define WS_NODE(k)  ((size_t)(k) * 131072)
#define WS_ACC_I    ((size_t)12 * 131072)          // acc_h_sc (i-side einsums)
#define WS_ACC_J    (WS_ACC_I + 131072)            // acc_h_st (j-side einsums)
#define WS_SUM      (WS_ACC_J + 131072)            // 5 tensors x 256 ch
#define WS_SSQ      (WS_SUM + 1280)
#define WS_MEAN     (WS_SSQ + 1280)
#define WS_RSTD     (WS_MEAN + 1280)
#define WS_H2       (WS_RSTD + 1280)               // 2 x 131072 (h2 pre-BN)
#define WS_ZERO_BEGIN WS_ACC_I
#define WS_ZERO_COUNT (131072 * 2 + 2560)

// ---- d_out layout (flat, return order) -------------------------------------
#define OUT_HSC 0
#define OUT_HST ((size_t)131072)
#define OUT_BI  ((size_t)262144)
#define OUT_SC  ((size_t)262144 + (size_t)33554432)
#define OUT_ST  ((size_t)262144 + (size_t)2 * 33554432)

// D = A(16x4 f32) * B(4x16 f32) + C(16x16 f32)
__device__ __forceinline__ v8f wmma_f32(v2f a, v2f b, v8f c) {
    return __builtin_amdgcn_wmma_f32_16x16x4_f32(
        /*neg_a=*/false, a, /*neg_b=*/false, b,
        /*c_mod=*/(short)0, c, /*reuse_a=*/false, /*reuse_b=*/false);
}

// ---------------------------------------------------------------------------
__global__ void k_zero(float* __restrict__ ws) {
    size_t i = (size_t)blockIdx.x * blockDim.x + threadIdx.x;
    if (i < (size_t)WS_ZERO_COUNT) ws[WS_ZERO_BEGIN + i] = 0.f;
}

// ---------------------------------------------------------------------------
// 12 node linears: out[r, n] = sum_k X[r,k] * W[slot][n,k] + bs[slot][n]
// grid = (32 row-tiles, 12 slots), block = 256 (8 waves). Wave handles 2 n-tiles.
__global__ void k_node_lin(const float* __restrict__ h_sc,
                           const float* __restrict__ h_st,
                           const float* __restrict__ Ws,
                           const float* __restrict__ bs,
                           float* __restrict__ ws) {
    int s  = blockIdx.y;                       // 0..11 (dest buffer)
    int rt = blockIdx.x;                       // 0..31 (row tile)
    int wslot  = (s < 8) ? s : (s < 10 ? s + 1 : s + 2);   // Ws index
    int srcsel = (s < 8) ? (s & 1) : (s >= 10);            // 0:h_sc 1:h_st

    int wave = threadIdx.x >> 5;
    int lane = threadIdx.x & 31;
    int mm   = lane & 15;
    int half = lane >> 4;
    int kh   = half << 1;

    const float* X    = srcsel ? h_st : h_sc;
    const float* W    = Ws + (size_t)wslot * WH * WH;
    const float* bias = bs + (size_t)wslot * WH;
    float*       dst  = ws + WS_NODE(s);

    const float* arow = X + (size_t)(rt * 16 + mm) * WH + kh;
    const float* brow = W + (size_t)(wave * 32 + mm) * WH + kh;  // n-tiles wave*2, wave*2+1

    v8f acc0 = {}, acc1 = {};
    #pragma unroll 8
    for (int k0 = 0; k0 < WH; k0 += 4) {
        v2f a  = *(const v2f*)(arow + k0);
        v2f b0 = *(const v2f*)(brow + k0);
        v2f b1 = *(const v2f*)(brow + k0 + 16 * WH);
        acc0 = wmma_f32(a, b0, acc0);
        acc1 = wmma_f32(a, b1, acc1);
    }
    #pragma unroll
    for (int g = 0; g < 2; ++g) {
        int ch = (wave * 2 + g) * 16 + mm;
        float bv = bias[ch];
        v8f acc = g ? acc1 : acc0;
        #pragma unroll
        for (int v = 0; v < 8; ++v) {
            int row = rt * 16 + v + 8 * half;
            dst[(size_t)row * WH + ch] = acc[v] + bv;
        }
    }
}

// ---------------------------------------------------------------------------
// Edge kernels. grid = (16 j-tiles, 16 i-tiles, 3 edges * B), block = 256.
// Each wave computes 2 row-tiles x 4 n-tiles per n-group (register blocked):
// per 8 WMMAs the K-loop fetches only 2 A-frags + 4 B-frags.
// MODE 0: e2 = e@Wc^T + bias + A[i] + B[j]; BN stats; sigmoid-gated einsums.
// MODE 1: recompute e2; BN normalize + relu + residual; write d_out.
template <int MODE>
__global__ void k_edge(const float* __restrict__ bi_e,
                       const float* __restrict__ sc_e,
                       const float* __restrict__ st_e,
                       const float* __restrict__ Ws,
                       const float* __restrict__ bs,
                       const float* __restrict__ gamma_e,
                       const float* __restrict__ beta_e,
                       float* __restrict__ ws,
                       float* __restrict__ out) {
    extern __shared__ float smem[];            // MODE 0 only: 256+256+4096 floats
    float* s_sum  = smem;
    float* s_ssq  = smem + WH;
    float* s_jacc = smem + 2 * WH;

    int z    = blockIdx.z;
    int eidx = z >> 1;           // 0 bi, 1 sc, 2 st
    int b    = z & 1;
    int i0   = blockIdx.y * 16;
    int j0   = blockIdx.x * 16;

    const float* e_src = (eidx == 0) ? bi_e : (eidx == 1) ? sc_e : st_e;
    int cslot          = (eidx == 0) ? 8 : (eidx == 1) ? 11 : 14;
    const float* Aterm = ws + WS_NODE((eidx == 0) ? 6 : (eidx == 1) ? 8 : 10);
    const float* Bterm = ws + WS_NODE((eidx == 0) ? 7 : (eidx == 1) ? 9 : 11);
    const int do_i = (eidx != 2);   // bi: Vh_st -> acc_h_sc ; sc: Wh_sc -> acc_h_sc
    const int do_j = (eidx != 1);   // bi: Vh_sc -> acc_h_st ; st: Wh_st -> acc_h_st
    const float* nbr_i = (eidx == 0) ? (ws + WS_NODE(3)) : (ws + WS_NODE(4));
    const float* nbr_j = (eidx == 0) ? (ws + WS_NODE(2)) : (ws + WS_NODE(5));
    float* acc_i = ws + WS_ACC_I;
    float* acc_j = ws + WS_ACC_J;
    float* gsum  = ws + WS_SUM + (size_t)eidx * WH;
    float* gssq  = ws + WS_SSQ + (size_t)eidx * WH;
    const float* meanp = ws + WS_MEAN + (size_t)eidx * WH;
    const float* rstdp = ws + WS_RSTD + (size_t)eidx * WH;
    float* eout = out + ((eidx == 0) ? OUT_BI : (eidx == 1) ? OUT_SC : OUT_ST);

    const float* W     = Ws + (size_t)cslot * WH * WH;
    const float* biasC = bs + (size_t)cslot * WH;

    if (MODE == 0) {
        for (int t = threadIdx.x; t < WH; t += blockDim.x) { s_sum[t] = 0.f; s_ssq[t] = 0.f; }
        for (int t = threadIdx.x; t < 16 * WH; t += blockDim.x) s_jacc[t] = 0.f;
        __syncthreads();
    }

    int wave = threadIdx.x >> 5;
    int lane = threadIdx.x & 31;
    int mm   = lane & 15;
    int half = lane >> 4;
    int kh   = half << 1;

    int iA = i0 + wave;                        // row-tile pair for this wave
    int iB = i0 + wave + 8;
    const float* arowA = e_src + (((size_t)b * WV + iA) * WV + j0 + mm) * WH + kh;
    const float* browB = W + (size_t)mm * WH + kh;

    #pragma unroll 1
    for (int ng = 0; ng < 4; ++ng) {           // n-tile groups of 4
        v8f acc[2][4] = {};
        const float* brow = browB + (size_t)ng * 64 * WH;
        #pragma unroll 4
        for (int k0 = 0; k0 < WH; k0 += 4) {
            v2f aA = *(const v2f*)(arowA + k0);
            v2f aB = *(const v2f*)(arowA + k0 + (size_t)8 * WV * WH);
            #pragma unroll
            for (int q = 0; q < 4; ++q) {
                v2f bb = *(const v2f*)(brow + k0 + (size_t)q * 16 * WH);
                acc[0][q] = wmma_f32(aA, bb, acc[0][q]);
                acc[1][q] = wmma_f32(aB, bb, acc[1][q]);
            }
        }
        // ---- epilogue for the 2x4 tile block ----
        #pragma unroll
        for (int g = 0; g < 2; ++g) {
            int i = g ? iB : iA;
            #pragma unroll 1
            for (int q = 0; q < 4; ++q) {
                int ch = (ng * 4 + q) * 16 + mm;
                float at = Aterm[((size_t)b * WV + i) * WH + ch];
                float bc = biasC[ch];

                float e2v[8];
                #pragma unroll
                for (int v = 0; v < 8; ++v) {
                    int j = j0 + v + 8 * half;
                    float bt = Bterm[((size_t)b * WV + j) * WH + ch];
                    e2v[v] = acc[g][q][v] + bc + at + bt;
                }

                if (MODE == 0) {
                    float ls = 0.f, lq = 0.f, isum = 0.f;
                    float nj = do_j ? nbr_j[((size_t)b * WV + i) * WH + ch] : 0.f;
                    #pragma unroll
                    for (int v = 0; v < 8; ++v) {
                        int j = j0 + v + 8 * half;
                        float e2 = e2v[v];
                        ls += e2;
                        lq += e2 * e2;
                        float gg = 1.f / (1.f + __expf(-e2));
                        if (do_i) isum += gg * nbr_i[((size_t)b * WV + j) * WH + ch];
                        if (do_j) atomicAdd(&s_jacc[(v + 8 * half) * WH + ch], gg * nj);
                    }
                    atomicAdd(&s_sum[ch], ls);
                    atomicAdd(&s_ssq[ch], lq);
                    if (do_i) {
                        isum += __shfl_xor(isum, 16, 32);   // combine j 0-7 with 8-15
                        if (half == 0)
                            atomicAdd(&acc_i[((size_t)b * WV + i) * WH + ch], isum);
                    }
                } else {
                    float mu = meanp[ch], rs = rstdp[ch];
                    float ga = gamma_e[ch], be = beta_e[ch];
                    #pragma unroll
                    for (int v = 0; v < 8; ++v) {
                        int j = j0 + v + 8 * half;
                        size_t idx = (((size_t)b * WV + i) * WV + j) * WH + ch;
                        float y = ga * (e2v[v] - mu) * rs + be;
                        __builtin_nontemporal_store(
                            e_src[idx] + (y > 0.f ? y : 0.f), &eout[idx]);
                    }
                }
            }
        }
    }

    if (MODE == 0) {
        __syncthreads();
        for (int t = threadIdx.x; t < WH; t += blockDim.x) {
            atomicAdd(&gsum[t], s_sum[t]);
            atomicAdd(&gssq[t], s_ssq[t]);
        }
        if (do_j) {
            for (int t = threadIdx.x; t < 16 * WH; t += blockDim.x) {
                int jl = t >> 8, ch = t & 255;
                atomicAdd(&acc_j[((size_t)b * WV + j0 + jl) * WH + ch], s_jacc[t]);
            }
        }
    }
}

// ---------------------------------------------------------------------------
// h2 = Uh + aggregated einsums; BN stats for h tensors (idx 3,4).
// grid = (16 row-chunks of 32, 2 tensors), block = 256 (thread == channel).
__global__ void k_h1(float* __restrict__ ws) {
    int t  = blockIdx.y;
    int ch = threadIdx.x;
    int r0 = blockIdx.x * 32;
    const float* U   = ws + WS_NODE(t);                      // 0: Uh_sc, 1: Uh_st
    const float* acc = ws + (t == 0 ? WS_ACC_I : WS_ACC_J);
    float* h2 = ws + WS_H2 + (size_t)t * 131072;
    float s = 0.f, q = 0.f;
    for (int r = r0; r < r0 + 32; ++r) {
        float v = U[(size_t)r * WH + ch] + acc[(size_t)r * WH + ch];
        h2[(size_t)r * WH + ch] = v;
        s += v; q += v * v;
    }
    atomicAdd(&ws[WS_SUM + (size_t)(3 + t) * WH + ch], s);
    atomicAdd(&ws[WS_SSQ + (size_t)(3 + t) * WH + ch], q);
}

// mean / rstd for all 5 tensors. grid = 5, block = 256.
__global__ void k_stats(float* __restrict__ ws) {
    int t = blockIdx.x, ch = threadIdx.x;
    float N   = (t < 3) ? (float)EDGE_ROWS : (float)NROWS_H;
    float s   = ws[WS_SUM + (size_t)t * WH + ch];
    float q   = ws[WS_SSQ + (size_t)t * WH + ch];
    float mu  = s / N;
    float var = q / N - mu * mu;
    ws[WS_MEAN + (size_t)t * WH + ch] = mu;
    ws[WS_RSTD + (size_t)t * WH + ch] = rsqrtf(var + EPS_BN);
}

// node outputs: out = h_in + relu(BN(h2)). grid = (512 rows, 2), block = 256.
__global__ void k_hfin(const float* __restrict__ h_sc,
                       const float* __restrict__ h_st,
                       const float* __restrict__ gamma_h,
                       const float* __restrict__ beta_h,
                       const float* __restrict__ ws,
                       float* __restrict__ out) {
    int t = blockIdx.y, r = blockIdx.x, ch = threadIdx.x;
    const float* h2  = ws + WS_H2 + (size_t)t * 131072;
    const float* hin = (t == 0) ? h_sc : h_st;
    float mu = ws[WS_MEAN + (size_t)(3 + t) * WH + ch];
    float rs = ws[WS_RSTD + (size_t)(3 + t) * WH + ch];
    float y  = gamma_h[ch] * (h2[(size_t)r * WH + ch] - mu) * rs + beta_h[ch];
    out[(size_t)t * 131072 + (size_t)r * WH + ch] =
        hin[(size_t)r * WH + ch] + (y > 0.f ? y : 0.f);
}

// ---------------------------------------------------------------------------
extern "C" void kernel_launch(void* const* d_in, const int* in_sizes, int n_in,
                              void* d_out, int out_size, void* d_ws, size_t ws_size,
                              hipStream_t stream) {
    const float* h_sc    = (const float*)d_in[0];
    const float* h_st    = (const float*)d_in[1];
    const float* bi_e    = (const float*)d_in[2];
    const float* sc_e    = (const float*)d_in[4];
    const float* st_e    = (const float*)d_in[6];
    const float* Ws      = (const float*)d_in[8];
    const float* bs      = (const float*)d_in[9];
    const float* gamma_h = (const float*)d_in[10];
    const float* beta_h  = (const float*)d_in[11];
    const float* gamma_e = (const float*)d_in[12];
    const float* beta_e  = (const float*)d_in[13];
    float* ws  = (float*)d_ws;
    float* out = (float*)d_out;

    const size_t lds_p1 = (size_t)(2 * WH + 16 * WH) * sizeof(float);  // 18 KB

    k_zero<<<dim3((WS_ZERO_COUNT + 255) / 256), 256, 0, stream>>>(ws);
    k_node_lin<<<dim3(32, 12), 256, 0, stream>>>(h_sc, h_st, Ws, bs, ws);
    k_edge<0><<<dim3(16, 16, 6), 256, lds_p1, stream>>>(bi_e, sc_e, st_e, Ws, bs,
                                                        gamma_e, beta_e, ws, out);
    k_h1<<<dim3(16, 2), 256, 0, stream>>>(ws);
    k_stats<<<dim3(5), 256, 0, stream>>>(ws);
    k_hfin<<<dim3(512, 2), 256, 0, stream>>>(h_sc, h_st, gamma_h, beta_h, ws, out);
    k_edge<1><<<dim3(16, 16, 6), 256, 0, stream>>>(bi_e, sc_e, st_e, Ws, bs,
                                                   gamma_e, beta_e, ws, out);
}